// GeodesicLoss_12051678232929
// MI455X (gfx1250) — compile-verified
//
#include <hip/hip_runtime.h>
#include <math.h>

typedef float v2f __attribute__((ext_vector_type(2)));
typedef float v8f __attribute__((ext_vector_type(8)));

#define NSTEPS 10
#define EPSF   1e-8f
#define BB     64
#define DD     128
#define BD     (BB*DD)        /* 8192  */
#define D2     (DD*DD)        /* 16384 */
#define D3     (D2*DD)        /* 2097152 */

// ---------------------------------------------------------------------------
// contract: acc[b,i] = -sum_{j,k} gamma[b,i,j,k] * v[b,j] * v[b,k]
// One block = one (b, 16-row i-tile). 8 waves split the flattened 16384-wide
// (j,k) axis; each wave runs V_WMMA_F32_16X16X4_F32 with A = 16x4 gamma chunk
// and B = w[kk..kk+3] replicated across all 16 columns (w = v_j * v_k).
// gamma is streamed with non-temporal hints (512 MB > L2, zero reuse), and
// two independent accumulators break the WMMA RAW chain.
// ---------------------------------------------------------------------------
__global__ void __launch_bounds__(256)
contract_kernel(const float* __restrict__ gamma,
                const float* __restrict__ v,
                float* __restrict__ acc) {
    __shared__ float s_v[DD];
    __shared__ float s_part[8 * 16];

    const int blk   = blockIdx.x;
    const int b     = blk >> 3;       // batch index
    const int itile = blk & 7;        // which 16-row tile of i
    const int tid   = threadIdx.x;
    const int wave  = tid >> 5;
    const int lane  = tid & 31;

    if (tid < DD) s_v[tid] = v[b * DD + tid];
    __syncthreads();

    const int row  = itile * 16 + (lane & 15);   // i row this lane feeds
    const int half = lane >> 4;                  // K half: lanes 0-15 -> K{0,1}, 16-31 -> K{2,3}
    const float* __restrict__ grow =
        gamma + (size_t)b * (size_t)D3 + (size_t)row * (size_t)D2;

    v8f c0 = {0.f, 0.f, 0.f, 0.f, 0.f, 0.f, 0.f, 0.f};
    v8f c1 = {0.f, 0.f, 0.f, 0.f, 0.f, 0.f, 0.f, 0.f};

    const int kk0 = wave * (D2 / 8);
    const int kk1 = kk0 + (D2 / 8);
    for (int kk = kk0; kk < kk1; kk += 8) {
        // ---- WMMA #0: K window [kk, kk+4) ----
        // A operand: lane (row, half) holds gamma[row, kk + 2*half + {0,1}]
        v2f a0 = __builtin_nontemporal_load((const v2f*)(grow + kk + 2 * half));
        int i0 = kk + 2 * half;
        v2f b0;
        b0.x = s_v[i0 >> 7] * s_v[i0 & 127];
        b0.y = s_v[(i0 + 1) >> 7] * s_v[(i0 + 1) & 127];
        c0 = __builtin_amdgcn_wmma_f32_16x16x4_f32(false, a0, false, b0,
                                                   (short)0, c0, false, false);

        // ---- WMMA #1: K window [kk+4, kk+8) (independent accumulator) ----
        v2f a1 = __builtin_nontemporal_load((const v2f*)(grow + kk + 4 + 2 * half));
        int i1 = kk + 4 + 2 * half;
        v2f b1;
        b1.x = s_v[i1 >> 7] * s_v[i1 & 127];
        b1.y = s_v[(i1 + 1) >> 7] * s_v[(i1 + 1) & 127];
        c1 = __builtin_amdgcn_wmma_f32_16x16x4_f32(false, a1, false, b1,
                                                   (short)0, c1, false, false);
    }

    v8f c = c0 + c1;

    // D tile column N=0: lane 0 holds rows 0..7 in c[0..7]; lane 16 rows 8..15.
    if (lane == 0) {
#pragma unroll
        for (int r = 0; r < 8; ++r) s_part[wave * 16 + r] = c[r];
    }
    if (lane == 16) {
#pragma unroll
        for (int r = 0; r < 8; ++r) s_part[wave * 16 + 8 + r] = c[r];
    }
    __syncthreads();

    if (tid < 16) {
        float s = 0.f;
#pragma unroll
        for (int w = 0; w < 8; ++w) s += s_part[w * 16 + tid];
        acc[b * DD + itile * 16 + tid] = -s;   // minus sign from reference
    }
}

// ---------------------------------------------------------------------------
// init: v = normalize((x1-x0)/NSTEPS) per row (+eps), traj = path[0] = x0
// ---------------------------------------------------------------------------
__global__ void init_kernel(const float* __restrict__ x0,
                            const float* __restrict__ x1,
                            float* __restrict__ v,
                            float* __restrict__ traj,
                            float* __restrict__ path0) {
    __shared__ float red[DD];
    const int b = blockIdx.x, d = threadIdx.x;
    const float o = x0[b * DD + d];
    const float t = x1[b * DD + d];
    const float diff = (t - o) * (1.0f / NSTEPS);
    red[d] = diff * diff;
    __syncthreads();
    for (int s = DD / 2; s > 0; s >>= 1) {
        if (d < s) red[d] += red[d + s];
        __syncthreads();
    }
    const float nrm = sqrtf(red[0]) + EPSF;
    v[b * DD + d]     = diff / nrm;
    traj[b * DD + d]  = o;
    path0[b * DD + d] = o;
}

// mid = v + scale * acc
__global__ void axpy_kernel(const float* __restrict__ v,
                            const float* __restrict__ acc,
                            float* __restrict__ mid, float scale) {
    const int i = blockIdx.x * blockDim.x + threadIdx.x;
    if (i < BD) mid[i] = v[i] + scale * acc[i];
}

// v += macc*dt; traj += v*dt; path_s1 = traj
__global__ void update_kernel(float* __restrict__ v,
                              const float* __restrict__ macc,
                              float* __restrict__ traj,
                              float* __restrict__ path_s1, float dt) {
    const int i = blockIdx.x * blockDim.x + threadIdx.x;
    if (i < BD) {
        const float nv = v[i] + macc[i] * dt;
        v[i] = nv;
        const float nt = traj[i] + nv * dt;
        traj[i] = nt;
        path_s1[i] = nt;
    }
}

// ---------------------------------------------------------------------------
// norms: blocks 0..63           -> ||traj[b,:] - x1[b,:]||
//        blocks 64..64+9*64-1   -> ||path[t+2]-2*path[t+1]+path[t]|| per (t,b)
// ---------------------------------------------------------------------------
__global__ void norm_kernel(const float* __restrict__ traj,
                            const float* __restrict__ x1,
                            const float* __restrict__ path,
                            float* __restrict__ norms) {
    __shared__ float red[DD];
    const int m = blockIdx.x, d = threadIdx.x;
    float val;
    if (m < BB) {
        val = traj[m * DD + d] - x1[m * DD + d];
    } else {
        const int m2 = m - BB;
        const int t  = m2 / BB;
        const int b  = m2 % BB;
        const float* p0 = path + (size_t)t * BD + b * DD + d;
        val = p0[2 * BD] - 2.0f * p0[BD] + p0[0];
    }
    red[d] = val * val;
    __syncthreads();
    for (int s = DD / 2; s > 0; s >>= 1) {
        if (d < s) red[d] += red[d + s];
        __syncthreads();
    }
    if (d == 0) norms[m] = sqrtf(red[0]);
}

// loss = mean_b ||traj-x1|| + 0.1 * sum_t mean_b ||sd||  =  (S1 + 0.1*S2)/64
__global__ void final_kernel(const float* __restrict__ norms,
                             float* __restrict__ out) {
    __shared__ float red[256];
    const int tid = threadIdx.x;
    const int n = BB + (NSTEPS - 1) * BB;   // 640
    float s = 0.f;
    for (int i = tid; i < n; i += 256) {
        const float w = (i < BB) ? 1.0f : 0.1f;
        s += w * norms[i];
    }
    red[tid] = s;
    __syncthreads();
    for (int st = 128; st > 0; st >>= 1) {
        if (tid < st) red[tid] += red[tid + st];
        __syncthreads();
    }
    if (tid == 0) out[0] = red[0] / (float)BB;
}

// ---------------------------------------------------------------------------
extern "C" void kernel_launch(void* const* d_in, const int* in_sizes, int n_in,
                              void* d_out, int out_size, void* d_ws, size_t ws_size,
                              hipStream_t stream) {
    (void)in_sizes; (void)n_in; (void)out_size; (void)ws_size;
    const float* x0    = (const float*)d_in[0];   // outputs [B,D]
    const float* x1    = (const float*)d_in[1];   // targets [B,D]
    const float* gamma = (const float*)d_in[2];   // gamma   [B,D,D,D]

    float* ws    = (float*)d_ws;
    float* v     = ws;                 // 8192
    float* acc   = ws + 1 * BD;        // 8192
    float* mid   = ws + 2 * BD;        // 8192
    float* macc  = ws + 3 * BD;        // 8192
    float* traj  = ws + 4 * BD;        // 8192
    float* path  = ws + 5 * BD;        // (NSTEPS+1)*8192
    float* norms = ws + 5 * BD + (NSTEPS + 1) * BD;  // 640

    const float dt = 1.0f / NSTEPS;

    init_kernel<<<BB, DD, 0, stream>>>(x0, x1, v, traj, path);

    for (int s = 0; s < NSTEPS; ++s) {
        contract_kernel<<<BB * (DD / 16), 256, 0, stream>>>(gamma, v, acc);
        axpy_kernel<<<BD / 256, 256, 0, stream>>>(v, acc, mid, 0.5f * dt);
        contract_kernel<<<BB * (DD / 16), 256, 0, stream>>>(gamma, mid, macc);
        update_kernel<<<BD / 256, 256, 0, stream>>>(v, macc, traj,
                                                    path + (size_t)(s + 1) * BD, dt);
    }

    norm_kernel<<<BB + (NSTEPS - 1) * BB, DD, 0, stream>>>(traj, x1, path, norms);
    final_kernel<<<1, 256, 0, stream>>>(norms, (float*)d_out);
}